// SubnetworkBlock_17763984736586
// MI455X (gfx1250) — compile-verified
//
#include <hip/hip_runtime.h>
#include <math.h>

typedef __attribute__((ext_vector_type(2))) float v2f;
typedef __attribute__((ext_vector_type(8))) float v8f;

#define SUBNETS 32
#define BATCH 131072
#define H1 10
#define H2 6
#define TILE 16
#define WAVES_PER_SUBNET 256
#define TILES_PER_WAVE 32          // 8192 tiles per subnet / 256 waves
#define WAVES_PER_BLOCK 8
#define ABLOCKS 1024               // 8192 waves total = 32 subnets * 256 waves
#define SMOOTH_LAMBDA 0.001f
#define BN_EPS 1e-10f

// ws layout
#define WS_PARTIALS_OFF 0                      // 32*256*3 floats = 98304 B
#define WS_STATS_OFF    98304                  // 32*3 floats
#define WS_XT_OFF       131072                 // 16 MB transposed x

__device__ __forceinline__ float fast_tanh(float x) {
#if __has_builtin(__builtin_amdgcn_tanhf)
  return __builtin_amdgcn_tanhf(x);
#elif __has_builtin(__builtin_amdgcn_tanh_f32)
  return __builtin_amdgcn_tanh_f32(x);
#else
  return tanhf(x);
#endif
}

// ---- 8-lane xor-butterfly add via DPP8 (pure VALU, no LDS, no waits) ------
#define DPP8_SEL(a,b,c,d,e,f,g,h) \
  ((a)|((b)<<3)|((c)<<6)|((d)<<9)|((e)<<12)|((f)<<15)|((g)<<18)|((h)<<21))

__device__ __forceinline__ float dpp8_xor_add1(float v) {
#if __has_builtin(__builtin_amdgcn_mov_dpp8)
  int t = __builtin_amdgcn_mov_dpp8(__builtin_bit_cast(int, v),
                                    DPP8_SEL(1,0,3,2,5,4,7,6));
  return v + __builtin_bit_cast(float, t);
#else
  return v + __shfl_xor(v, 1, 32);
#endif
}
__device__ __forceinline__ float dpp8_xor_add2(float v) {
#if __has_builtin(__builtin_amdgcn_mov_dpp8)
  int t = __builtin_amdgcn_mov_dpp8(__builtin_bit_cast(int, v),
                                    DPP8_SEL(2,3,0,1,6,7,4,5));
  return v + __builtin_bit_cast(float, t);
#else
  return v + __shfl_xor(v, 2, 32);
#endif
}
__device__ __forceinline__ float dpp8_xor_add4(float v) {
#if __has_builtin(__builtin_amdgcn_mov_dpp8)
  int t = __builtin_amdgcn_mov_dpp8(__builtin_bit_cast(int, v),
                                    DPP8_SEL(4,5,6,7,0,1,2,3));
  return v + __builtin_bit_cast(float, t);
#else
  return v + __shfl_xor(v, 4, 32);
#endif
}
// sum over the 8-lane group (sufficient: all nonzero N-contributions and all
// consuming lanes live in lanes 0..7 of each 16-half)
__device__ __forceinline__ float reduce8(float v) {
  return dpp8_xor_add4(dpp8_xor_add2(dpp8_xor_add1(v)));
}

// ---------------------------------------------------------------------------
// Kernel 0: 32x32 tiled transpose  x[BATCH][32] -> xT[32][BATCH]
// ---------------------------------------------------------------------------
__global__ __launch_bounds__(256) void transpose_x(const float* __restrict__ x,
                                                   float* __restrict__ xT) {
  __shared__ float tileS[32][33];
  const int b0  = blockIdx.x * 32;
  const int col = threadIdx.x & 31;
  const int row = threadIdx.x >> 5;      // 0..7
#pragma unroll
  for (int i = 0; i < 4; ++i) {
    const int r = row + i * 8;
    tileS[r][col] = x[(size_t)(b0 + r) * SUBNETS + col];
  }
  __syncthreads();
#pragma unroll
  for (int i = 0; i < 4; ++i) {
    const int s = row + i * 8;           // subnet index
    xT[(size_t)s * BATCH + b0 + col] = tileS[col][s];
  }
}

// ---------------------------------------------------------------------------
// Kernel 1: fused MLP fwd + analytic d2/dx2 via three fp32 WMMA matmuls.
// One wave handles one subnet, 32 tiles of 16 batch rows each.
// ---------------------------------------------------------------------------
__global__ __launch_bounds__(256) void subnet_fused(
    const float* __restrict__ xsrc, int transposed,
    const float* __restrict__ W1, const float* __restrict__ b1,
    const float* __restrict__ W2, const float* __restrict__ b2,
    const float* __restrict__ W3, const float* __restrict__ b3,
    float* __restrict__ outRaw,          // d_out: raw pre-BN outputs [BATCH][32]
    float* __restrict__ partials)        // [32][256][3]
{
  const int tid   = threadIdx.x;
  const int lane  = tid & 31;
  const int wave  = tid >> 5;
  const int half  = lane >> 4;           // which 16-lane half
  const int mrow  = lane & 15;           // A-side row m / B-side col n
  const int gwave = blockIdx.x * WAVES_PER_BLOCK + wave;   // 0..8191
  const int s     = gwave >> 8;          // subnet (256 waves per subnet)
  const int wIdx  = gwave & 255;

  // --- per-lane parameter fragments (padding lanes get zeros) -------------
  float w1A[3][2], b1A[3][2], w1A2[3][2];
  v2f   Bv[3];
#pragma unroll
  for (int c = 0; c < 3; ++c) {
#pragma unroll
    for (int j = 0; j < 2; ++j) {
      const int  k  = 4 * c + 2 * half + j;   // A fragment K index for this lane
      const bool ok = (k < H1);
      w1A[c][j]  = ok ? W1[s * H1 + k] : 0.f;
      b1A[c][j]  = ok ? b1[s * H1 + k] : 0.f;
      w1A2[c][j] = w1A[c][j] * w1A[c][j];
    }
    const int nB  = mrow;
    const int kb0 = 4 * c + half;
    const int kb1 = 4 * c + 2 + half;
    Bv[c].x = (kb0 < H1 && nB < H2) ? W2[(s * H1 + kb0) * H2 + nB] : 0.f;
    Bv[c].y = (kb1 < H1 && nB < H2) ? W2[(s * H1 + kb1) * H2 + nB] : 0.f;
  }
  const int   n   = mrow;
  const float b2n = (n < H2) ? b2[s * H2 + n] : 0.f;
  const float w3n = (n < H2) ? W3[s * H2 + n] : 0.f;
  const float b3s = b3[s];

  float acc1 = 0.f, acc2 = 0.f, accg = 0.f;

  // software-pipelined x load (one scalar per lane per tile)
  auto xAddr = [&](int t) -> const float* {
    const int b0 = (wIdx + t * WAVES_PER_SUBNET) * TILE;
    return transposed ? (xsrc + (size_t)s * BATCH + b0 + mrow)
                      : (xsrc + (size_t)(b0 + mrow) * SUBNETS + s);
  };
  float xm = *xAddr(0);

  for (int t = 0; t < TILES_PER_WAVE; ++t) {
    const int b0 = (wIdx + t * WAVES_PER_SUBNET) * TILE;
    const float xnext = (t + 1 < TILES_PER_WAVE) ? *xAddr(t + 1) : 0.f;

    v8f accH  = {0.f,0.f,0.f,0.f,0.f,0.f,0.f,0.f};
    v8f accD  = {0.f,0.f,0.f,0.f,0.f,0.f,0.f,0.f};
    v8f accD2 = {0.f,0.f,0.f,0.f,0.f,0.f,0.f,0.f};

#pragma unroll
    for (int c = 0; c < 3; ++c) {
      v2f Ah, Ad, Ad2;
#pragma unroll
      for (int j = 0; j < 2; ++j) {
        // layer 1 is elementwise in k: each lane builds its own A slots
        const float u  = fast_tanh(xm * w1A[c][j] + b1A[c][j]);
        const float sq = 1.f - u * u;
        const float d1 = sq * w1A[c][j];                 // dh1/dx
        const float d2 = -2.f * u * sq * w1A2[c][j];     // d2h1/dx2
        if (j == 0) { Ah.x = u; Ad.x = d1; Ad2.x = d2; }
        else        { Ah.y = u; Ad.y = d1; Ad2.y = d2; }
      }
      // D = A(16x4) * B(4x16) + C, fp32 matrix pipe
      accH  = __builtin_amdgcn_wmma_f32_16x16x4_f32(false, Ah,  false, Bv[c],
                                                    (short)0, accH,  false, false);
      accD  = __builtin_amdgcn_wmma_f32_16x16x4_f32(false, Ad,  false, Bv[c],
                                                    (short)0, accD,  false, false);
      accD2 = __builtin_amdgcn_wmma_f32_16x16x4_f32(false, Ad2, false, Bv[c],
                                                    (short)0, accD2, false, false);
    }

#pragma unroll
    for (int r = 0; r < 8; ++r) {        // C/D row r holds M=r (lanes<16) / M=r+8
      const float a2  = accH[r] + b2n;
      const float h2t = fast_tanh(a2);
      const float sq  = 1.f - h2t * h2t;
      const float d1  = accD[r];
      const float hp  = sq * d1;                               // h2'
      const float hpp = sq * accD2[r] - 2.f * h2t * hp * d1;   // h2''
      // reduce over N with 3-step DPP8 butterfly (n=0..5 nonzero, all in 8-group)
      const float oc = reduce8(h2t * w3n);
      const float gc = reduce8(hpp * w3n);
      const float outv = oc + b3s;
      if (mrow == r) {                   // one lane per (row, half) owns result
        outRaw[(size_t)(b0 + r + 8 * half) * SUBNETS + s] = outv;
        acc1 += outv;
        acc2 += outv * outv;
        accg += gc * gc;
      }
    }
    xm = xnext;
  }

  // deterministic wave-level reduction of the three accumulators (once/wave)
#pragma unroll
  for (int off = 1; off < 32; off <<= 1) {
    acc1 += __shfl_xor(acc1, off, 32);
    acc2 += __shfl_xor(acc2, off, 32);
    accg += __shfl_xor(accg, off, 32);
  }
  if (lane == 0) {
    float* p = partials + ((size_t)s * WAVES_PER_SUBNET + wIdx) * 3;
    p[0] = acc1; p[1] = acc2; p[2] = accg;
  }
}

// ---------------------------------------------------------------------------
// Kernel 2: per-subnet stats (deterministic tree reduction of 256 partials)
// ---------------------------------------------------------------------------
__global__ __launch_bounds__(256) void reduce_stats(const float* __restrict__ partials,
                                                    float* __restrict__ stats) {
  __shared__ float s1[256], s2[256], s3[256];
  const int s = blockIdx.x, t = threadIdx.x;
  const float* p = partials + ((size_t)s * 256 + t) * 3;
  s1[t] = p[0]; s2[t] = p[1]; s3[t] = p[2];
  __syncthreads();
  for (int off = 128; off > 0; off >>= 1) {
    if (t < off) { s1[t] += s1[t + off]; s2[t] += s2[t + off]; s3[t] += s3[t + off]; }
    __syncthreads();
  }
  if (t == 0) {
    const float inv  = 1.0f / (float)BATCH;
    const float mean = s1[0] * inv;
    const float var  = s2[0] * inv - mean * mean;
    stats[s * 3 + 0] = mean;
    stats[s * 3 + 1] = rsqrtf(var + BN_EPS);
    stats[s * 3 + 2] = (s3[0] * inv) * rsqrtf(var);   // penalty_s
  }
}

// ---------------------------------------------------------------------------
// Kernel 3: in-place BatchNorm of d_out (float4) + loss scalar
// ---------------------------------------------------------------------------
__global__ __launch_bounds__(256) void normalize_out(float4* __restrict__ out4,
                                                     const float* __restrict__ stats,
                                                     float* __restrict__ loss_out) {
  const int i = blockIdx.x * 256 + threadIdx.x;     // over BATCH*32/4 elements
  const int sb = (i & 7) * 4;                       // base subnet of this float4
  float4 v = out4[i];
  v.x = (v.x - stats[(sb + 0) * 3]) * stats[(sb + 0) * 3 + 1];
  v.y = (v.y - stats[(sb + 1) * 3]) * stats[(sb + 1) * 3 + 1];
  v.z = (v.z - stats[(sb + 2) * 3]) * stats[(sb + 2) * 3 + 1];
  v.w = (v.w - stats[(sb + 3) * 3]) * stats[(sb + 3) * 3 + 1];
  out4[i] = v;
  if (i == 0) {
    float acc = 0.f;
    for (int s = 0; s < SUBNETS; ++s) acc += stats[s * 3 + 2];
    *loss_out = SMOOTH_LAMBDA * acc;
  }
}

// ---------------------------------------------------------------------------
extern "C" void kernel_launch(void* const* d_in, const int* in_sizes, int n_in,
                              void* d_out, int out_size, void* d_ws, size_t ws_size,
                              hipStream_t stream) {
  const float* x  = (const float*)d_in[0];
  const float* W1 = (const float*)d_in[1];
  const float* b1 = (const float*)d_in[2];
  const float* W2 = (const float*)d_in[3];
  const float* b2 = (const float*)d_in[4];
  const float* W3 = (const float*)d_in[5];
  const float* b3 = (const float*)d_in[6];
  float* out = (float*)d_out;

  char*  ws       = (char*)d_ws;
  float* partials = (float*)(ws + WS_PARTIALS_OFF);
  float* stats    = (float*)(ws + WS_STATS_OFF);
  float* xT       = (float*)(ws + WS_XT_OFF);

  const size_t needT = (size_t)WS_XT_OFF + (size_t)BATCH * SUBNETS * sizeof(float);
  const int useT = (ws_size >= needT) ? 1 : 0;

  if (useT) {
    transpose_x<<<BATCH / 32, 256, 0, stream>>>(x, xT);
  }
  subnet_fused<<<ABLOCKS, 256, 0, stream>>>(useT ? xT : x, useT,
                                            W1, b1, W2, b2, W3, b3,
                                            out, partials);
  reduce_stats<<<SUBNETS, 256, 0, stream>>>(partials, stats);
  normalize_out<<<(BATCH * SUBNETS / 4) / 256, 256, 0, stream>>>(
      (float4*)out, stats, out + (size_t)BATCH * SUBNETS);
}